// RotaryPositionEncode_76244259439162
// MI455X (gfx1250) — compile-verified
//
#include <hip/hip_runtime.h>
#include <cstdint>

// RoPE, fp32, x:[B=8, S=4096, D=1024], positions:[S] int.
// Bandwidth-bound: 256 MiB of streaming traffic -> ~11.5 us floor at 23.3 TB/s.
// Strategy:
//  - one block per sequence row s; each thread owns one float4 column (2 pairs)
//  - sin/cos computed ONCE per (s, column), reused across all 8 batches
//  - exact Cody-Waite range reduction + hardware v_sin_f32/v_cos_f32
//    (arguments bounded: |angle| <= 4095 rad -> n <= 652 exact in fp32)
//  - x streamed via gfx1250 async global->LDS b128 loads (ASYNCcnt), 2-deep
//    double buffer, wave-private slots (no barriers needed)
//  - results written with non-temporal b128 stores (single-use stream)

typedef float v4f __attribute__((ext_vector_type(4)));

constexpr int B  = 8;
constexpr int S  = 4096;
constexpr int D  = 1024;
constexpr int D4 = D / 4;  // 256 float4 columns per row -> blockDim.x
constexpr uint32_t BATCH_BYTES = (uint32_t)S * (uint32_t)D * 4u;   // 16 MiB
// -2 * log2(10000) / D
constexpr float NEG2_LOG2_THETA_OVER_D = -0.0259525632393544f;

__device__ __forceinline__ void async_load_b128(uint32_t lds_byte_addr,
                                                uint32_t vbyte_off,
                                                uint64_t sbase) {
  // GVS mode: mem_addr = SGPR64 + VGPR32 offset; LDS dest offset from VGPR.
  asm volatile("global_load_async_to_lds_b128 %0, %1, %2"
               :
               : "v"(lds_byte_addr), "v"(vbyte_off), "s"(sbase)
               : "memory");
}

// sin/cos of `a` (radians, |a| <= ~4100) via exact 2-term Cody-Waite
// reduction and the native transcendental units (input in revolutions).
__device__ __forceinline__ void fast_sincos(float a, float& sn, float& cs) {
  constexpr float INV_2PI   = 0.15915494309189535f;
  constexpr float TWO_PI_HI = 6.28125f;               // 8 significant bits
  constexpr float TWO_PI_LO = 1.9353071795864769e-3f; // 2*pi - TWO_PI_HI
  const float n = rintf(a * INV_2PI);        // |n| <= 652, exact integer
  float r = fmaf(n, -TWO_PI_HI, a);          // exact
  r = fmaf(n, -TWO_PI_LO, r);                // |r| <= pi, ~1e-7 error
  const float rev = r * INV_2PI;             // |rev| <= 0.5
  sn = __builtin_amdgcn_sinf(rev);           // v_sin_f32: sin(2*pi*rev)
  cs = __builtin_amdgcn_cosf(rev);           // v_cos_f32
}

__global__ __launch_bounds__(256) void rope_kernel(const float* __restrict__ x,
                                                   const int* __restrict__ pos,
                                                   float* __restrict__ out) {
  __shared__ v4f stage[2 * D4];  // 2 buffers x 256 x 16B = 8 KiB, wave-private slots

  const int tid = threadIdx.x;   // float4 column in row: 0..255
  const int s   = blockIdx.x;    // sequence position (one row per block)

  // Angles for this thread's two pairs k0=2*tid, k1=2*tid+1 (computed once,
  // reused for all 8 batch slices).
  const float p  = (float)pos[s];
  const float a0 = p * exp2f((float)(2 * tid)     * NEG2_LOG2_THETA_OVER_D);
  const float a1 = p * exp2f((float)(2 * tid + 1) * NEG2_LOG2_THETA_OVER_D);
  float s0, c0, s1, c1;
  fast_sincos(a0, s0, c0);
  fast_sincos(a1, s1, c1);

  // Low 32 bits of a generic shared pointer == LDS byte offset (ISA 10.2).
  const uint32_t lds0  = (uint32_t)(uintptr_t)&stage[tid];
  const uint32_t lds1  = (uint32_t)(uintptr_t)&stage[D4 + tid];
  const uint32_t voff0 = (uint32_t)(s * (D * 4) + tid * 16);
  const uint64_t xbase = (uint64_t)(uintptr_t)x;

  // Prologue: kick off batch 0.
  async_load_b128(lds0, voff0, xbase);

#pragma unroll
  for (int b = 0; b < B; ++b) {
    if (b + 1 < B) {
      // The buffer we are about to overwrite was consumed via ds_load last
      // iteration; drain DS before letting the async engine write it again.
      asm volatile("s_wait_dscnt 0x0" ::: "memory");
      async_load_b128(((b + 1) & 1) ? lds1 : lds0,
                      voff0 + (uint32_t)(b + 1) * BATCH_BYTES, xbase);
      // Two loads in flight; wait until the older one (batch b) has landed.
      asm volatile("s_wait_asynccnt 0x1" ::: "memory");
    } else {
      asm volatile("s_wait_asynccnt 0x0" ::: "memory");
    }

    v4f v = stage[(b & 1) * D4 + tid];  // ds_load_b128 of our own slot
    v4f r;
    r.x = c0 * v.x - s0 * v.y;
    r.y = s0 * v.x + c0 * v.y;
    r.z = c1 * v.z - s1 * v.w;
    r.w = s1 * v.z + c1 * v.w;

    v4f* dst = (v4f*)(out + ((size_t)b * S + (size_t)s) * (size_t)D +
                      (size_t)tid * 4);
    __builtin_nontemporal_store(r, dst);  // single-use stream: keep L2 clean
  }
}

extern "C" void kernel_launch(void* const* d_in, const int* in_sizes, int n_in,
                              void* d_out, int out_size, void* d_ws, size_t ws_size,
                              hipStream_t stream) {
  (void)in_sizes; (void)n_in; (void)d_ws; (void)ws_size; (void)out_size;
  const float* x   = (const float*)d_in[0];
  const int*   pos = (const int*)d_in[1];
  float*       out = (float*)d_out;
  rope_kernel<<<dim3(S), dim3(D4), 0, stream>>>(x, pos, out);
}